// CSGDN_32031866093637
// MI455X (gfx1250) — compile-verified
//
#include <hip/hip_runtime.h>
#include <math.h>

#define FEAT 64
#define NEG_SLOPE 0.2f

typedef __attribute__((ext_vector_type(2))) float v2f;
typedef __attribute__((ext_vector_type(8))) float v8f;

// ---------------------------------------------------------------------------
// h[l] = x @ W[l]  via V_WMMA_F32_16X16X4_F32, one wave32 per 16x16 C tile.
// grid = (ceil(N/16), 4 col tiles, 4 layers), block = 32 (one wave).
// ---------------------------------------------------------------------------
__global__ __launch_bounds__(32) void gat_gemm_wmma(
    const float* __restrict__ x, const float* __restrict__ W,
    float* __restrict__ h, int N) {
  const int rt = blockIdx.x, ct = blockIdx.y, l = blockIdx.z;
  const int lane = threadIdx.x;
  const int lm = lane & 15;   // M index (A) / N index (B,C)
  const int hi = lane >> 4;   // upper half-wave selector
  const int m0 = rt << 4, n0 = ct << 4;
  const float* __restrict__ Wl = W + (size_t)l * FEAT * FEAT;
  float* __restrict__ hl = h + (size_t)l * N * FEAT;

  int rowA = m0 + lm;
  if (rowA > N - 1) rowA = N - 1;   // clamp (N=50000 is a multiple of 16; never hit)

  v8f acc = {};
#pragma unroll
  for (int kb = 0; kb < 16; ++kb) {
    const int k0 = (kb << 2) + (hi << 1);      // K = 4*kb + 2*hi + r, r=0,1
    v2f a, b;
    a.x = x[(size_t)rowA * FEAT + k0];         // consecutive -> 64-bit load
    a.y = x[(size_t)rowA * FEAT + k0 + 1];
    b.x = Wl[(size_t)(k0)     * FEAT + n0 + lm];
    b.y = Wl[(size_t)(k0 + 1) * FEAT + n0 + lm];
    acc = __builtin_amdgcn_wmma_f32_16x16x4_f32(
        /*neg_a=*/false, a, /*neg_b=*/false, b,
        /*c_mod=*/(short)0, acc, /*reuse_a=*/false, /*reuse_b=*/false);
  }
#pragma unroll
  for (int r = 0; r < 8; ++r) {                // C[M=r+8*hi][N=lm]
    const int row = m0 + r + (hi << 3);
    if (row < N) hl[(size_t)row * FEAT + n0 + lm] = acc[r];
  }
}

// ---------------------------------------------------------------------------
// s[l][i] = h[l][i]·a_src[l],  d[l][i] = h[l][i]·a_dst[l]
// ---------------------------------------------------------------------------
__global__ void gat_scores(const float* __restrict__ h,
                           const float* __restrict__ a_src,
                           const float* __restrict__ a_dst,
                           float* __restrict__ s, float* __restrict__ dd, int N) {
  const int i = blockIdx.x * blockDim.x + threadIdx.x;
  const int l = blockIdx.y;
  if (i >= N) return;
  const float4* hr = (const float4*)(h + ((size_t)l * N + i) * FEAT);
  const float4* as = (const float4*)(a_src + (size_t)l * FEAT);
  const float4* ad = (const float4*)(a_dst + (size_t)l * FEAT);
  float sv = 0.f, dv = 0.f;
#pragma unroll
  for (int q = 0; q < FEAT / 4; ++q) {
    const float4 v = hr[q], A = as[q], B = ad[q];
    sv += v.x * A.x + v.y * A.y + v.z * A.z + v.w * A.w;
    dv += v.x * B.x + v.y * B.y + v.z * B.z + v.w * B.w;
  }
  s[(size_t)l * N + i]  = sv;
  dd[(size_t)l * N + i] = dv;
}

__global__ void gat_init_md(float* __restrict__ m, float* __restrict__ denom, int n) {
  const int i = blockIdx.x * blockDim.x + threadIdx.x;
  if (i < n) { m[i] = -__builtin_inff(); denom[i] = 0.f; }
}

__device__ __forceinline__ float lrelu(float v) { return v > 0.f ? v : NEG_SLOPE * v; }

// Relaxed agent-scope atomics -> native global_atomic_* single instructions.
__device__ __forceinline__ void atomAddF(float* addr, float val) {
  (void)__hip_atomic_fetch_add(addr, val, __ATOMIC_RELAXED, __HIP_MEMORY_SCOPE_AGENT);
}

// float atomic max via ordered int/uint bit trick (init must be -inf)
__device__ __forceinline__ void atomicMaxF(float* addr, float val) {
  if (val >= 0.f)
    (void)__hip_atomic_fetch_max((int*)addr, __float_as_int(val),
                                 __ATOMIC_RELAXED, __HIP_MEMORY_SCOPE_AGENT);
  else
    (void)__hip_atomic_fetch_min((unsigned int*)addr, __float_as_uint(val),
                                 __ATOMIC_RELAXED, __HIP_MEMORY_SCOPE_AGENT);
}

// ---------------------------------------------------------------------------
// Pass A: e = lrelu(s[src]+d[dst]) (masked self-edges -> -1e9); store e in pw;
//         segment max into m[dst]. k in [E, E+N) is the added self-loop edge.
// ---------------------------------------------------------------------------
__global__ void gat_edge_max(const int* __restrict__ eidx,
                             const float* __restrict__ s, const float* __restrict__ dd,
                             float* __restrict__ m, float* __restrict__ pw,
                             int E, int N) {
  const int k = blockIdx.x * blockDim.x + threadIdx.x;
  if (k >= E + N) return;
  int dst; float e;
  if (k < E) {
    const int src = eidx[k];
    dst = eidx[E + k];
    e = lrelu(s[src] + dd[dst]);
    if (src == dst) e = -1e9f;     // remove_self_loops (masked out of softmax)
  } else {
    dst = k - E;                    // add_self_loops
    e = lrelu(s[dst] + dd[dst]);
  }
  pw[k] = e;
  atomicMaxF(m + dst, e);
}

// Pass B: p = exp(e - m[dst]); store p in pw; segment sum into denom[dst].
__global__ void gat_edge_sum(const int* __restrict__ eidx, float* __restrict__ pw,
                             const float* __restrict__ m, float* __restrict__ denom,
                             int E, int N) {
  const int k = blockIdx.x * blockDim.x + threadIdx.x;
  if (k >= E + N) return;
  const int dst = (k < E) ? eidx[E + k] : (k - E);
  const float p = __expf(pw[k] - m[dst]);
  pw[k] = p;
  atomAddF(denom + dst, p);
}

// Pass B2: alpha = p / denom[dst], folded into pw so the hot aggregation
// pass never gathers denom (kills 16x redundant random loads per edge).
__global__ void gat_edge_alpha(const int* __restrict__ eidx, float* __restrict__ pw,
                               const float* __restrict__ denom, int E, int N) {
  const int k = blockIdx.x * blockDim.x + threadIdx.x;
  if (k >= E + N) return;
  const int dst = (k < E) ? eidx[E + k] : (k - E);
  pw[k] = pw[k] / denom[dst];
}

// Pass C: out[dst] += alpha * h[src]; 16 threads/edge, float4 each.
__global__ void gat_edge_agg(const int* __restrict__ eidx,
                             const float* __restrict__ pw,
                             const float* __restrict__ hl,
                             float* __restrict__ o, int E, int N) {
  const size_t u = (size_t)blockIdx.x * blockDim.x + threadIdx.x;
  const int k = (int)(u >> 4);
  if (k >= E + N) return;
  const int f = ((int)u & 15) << 2;
  int src, dst;
  if (k < E) { src = eidx[k]; dst = eidx[E + k]; }
  else       { src = dst = k - E; }
  const float alpha = pw[k];
  const float4 hv = *(const float4*)(hl + (size_t)src * FEAT + f);
  float* op = o + (size_t)dst * FEAT + f;
  atomAddF(op + 0, alpha * hv.x);
  atomAddF(op + 1, alpha * hv.y);
  atomAddF(op + 2, alpha * hv.z);
  atomAddF(op + 3, alpha * hv.w);
}

// out = relu(acc + bias[layer])
__global__ void gat_finalize(float* __restrict__ o, const float* __restrict__ b, int N) {
  const size_t u = (size_t)blockIdx.x * blockDim.x + threadIdx.x;
  if (u >= (size_t)N * FEAT) return;
  const float v = o[u] + b[u & (FEAT - 1)];
  o[u] = v > 0.f ? v : 0.f;
}

// ---------------------------------------------------------------------------
extern "C" void kernel_launch(void* const* d_in, const int* in_sizes, int n_in,
                              void* d_out, int out_size, void* d_ws, size_t ws_size,
                              hipStream_t stream) {
  const int N = in_sizes[8] / FEAT;   // x is [N, 64]
  const int E = in_sizes[0] / 2;      // edge_index is [2, E]

  const float* x     = (const float*)d_in[8];
  const float* W     = (const float*)d_in[9];
  const float* a_src = (const float*)d_in[10];
  const float* a_dst = (const float*)d_in[11];
  const float* bias  = (const float*)d_in[12];
  float* out = (float*)d_out;

  // workspace layout (floats)
  float* h     = (float*)d_ws;                 // 4*N*64
  float* s     = h + (size_t)4 * N * FEAT;     // 4*N
  float* dd    = s + (size_t)4 * N;            // 4*N
  float* m     = dd + (size_t)4 * N;           // 8*N
  float* denom = m + (size_t)8 * N;            // 8*N
  float* pw    = denom + (size_t)8 * N;        // E+N (reused per conv)

  // 1) h[l] = x @ W[l] for all 4 layers (WMMA f32)
  dim3 gGemm((N + 15) / 16, FEAT / 16, 4);
  gat_gemm_wmma<<<gGemm, dim3(32), 0, stream>>>(x, W, h, N);

  // 2) per-node attention scores for all 4 layers
  dim3 gSc((N + 255) / 256, 4);
  gat_scores<<<gSc, 256, 0, stream>>>(h, a_src, a_dst, s, dd, N);

  // 3) init accumulators
  (void)hipMemsetAsync(d_out, 0, (size_t)out_size * sizeof(float), stream);
  gat_init_md<<<(8 * N + 255) / 256, 256, 0, stream>>>(m, denom, 8 * N);

  // output order: tp_a, tp_b, dp_a, dp_b, tn_a, tn_b, dn_a, dn_b
  const int edge_of[8]  = {0, 1, 4, 5, 2, 3, 6, 7};
  const int layer_of[8] = {0, 0, 2, 2, 1, 1, 3, 3};
  const int T = E + N;
  const int gE = (T + 255) / 256;
  const unsigned gC = (unsigned)(((size_t)T * 16 + 255) / 256);
  const unsigned gF = (unsigned)(((size_t)N * FEAT + 255) / 256);

  for (int c = 0; c < 8; ++c) {
    const int* eidx = (const int*)d_in[edge_of[c]];
    const int l = layer_of[c];
    const float* sl = s + (size_t)l * N;
    const float* dl = dd + (size_t)l * N;
    const float* hl = h + (size_t)l * N * FEAT;
    float* mc = m + (size_t)c * N;
    float* dc = denom + (size_t)c * N;
    float* oc = out + (size_t)c * N * FEAT;

    gat_edge_max<<<gE, 256, 0, stream>>>(eidx, sl, dl, mc, pw, E, N);
    gat_edge_sum<<<gE, 256, 0, stream>>>(eidx, pw, mc, dc, E, N);
    gat_edge_alpha<<<gE, 256, 0, stream>>>(eidx, pw, dc, E, N);
    gat_edge_agg<<<gC, 256, 0, stream>>>(eidx, pw, hl, oc, E, N);
    gat_finalize<<<gF, 256, 0, stream>>>(oc, bias + (size_t)l * FEAT, N);
  }
}